// RNN_45561013076400
// MI455X (gfx1250) — compile-verified
//
#include <hip/hip_runtime.h>

typedef float v2f __attribute__((ext_vector_type(2)));
typedef float v8f __attribute__((ext_vector_type(8)));

#define BSZ  8192
#define TLEN 512
#define FIN  8
#define NLAY 5

// ---------------------------------------------------------------------------
// Kernel A: G0[b,t,n] = sum_k x[b,t,k] * W_ih0[n,k] + (b_ih0[n] + b_hh0[n])
// One wave32 per 16 rows (rows flattened as r = b*T + t).
// Uses V_WMMA_F32_16X16X4_F32 twice (K = 0..3, 4..7), N padded 4 -> 16.
// ---------------------------------------------------------------------------
__global__ __launch_bounds__(256) void lstm0_gates_wmma(
    const float* __restrict__ x,     // [B,T,F]
    const float* __restrict__ Wih0,  // [4,8]
    const float* __restrict__ bih0,  // [4]
    const float* __restrict__ bhh0,  // [4]
    float* __restrict__ G0)          // [B,T,4]
{
    const int lane = threadIdx.x & 31;
    const int wave = (blockIdx.x * blockDim.x + threadIdx.x) >> 5;  // tile id
    const long r0  = (long)wave * 16;      // first row of 16-row tile
    const int  m   = lane & 15;            // A: row / B,C: column
    const int  kh  = lane >> 4;            // lane-half selects K pair (ISA 7.12.2)

    // A fragment: (vgpr j, lane) = A[m][2*kh + j]
    const float* xr = x + (r0 + m) * FIN;
    v2f a0, a1;
    a0.x = xr[2 * kh + 0];      a0.y = xr[2 * kh + 1];       // K chunk 0..3
    a1.x = xr[4 + 2 * kh + 0];  a1.y = xr[4 + 2 * kh + 1];   // K chunk 4..7

    // B fragment: (vgpr j, lane) = B[2*kh + j][n],  B[k][n] = Wih0[n][k], n<4 else 0
    v2f b0 = {0.f, 0.f}, b1 = {0.f, 0.f};
    float bias = 0.f;
    if (m < 4) {
        b0.x = Wih0[m * FIN + 2 * kh + 0];
        b0.y = Wih0[m * FIN + 2 * kh + 1];
        b1.x = Wih0[m * FIN + 4 + 2 * kh + 0];
        b1.y = Wih0[m * FIN + 4 + 2 * kh + 1];
        bias = bih0[m] + bhh0[m];
    }

    v8f c = {};
    c = __builtin_amdgcn_wmma_f32_16x16x4_f32(false, a0, false, b0,
                                              (short)0, c, false, false);
    c = __builtin_amdgcn_wmma_f32_16x16x4_f32(false, a1, false, b1,
                                              (short)0, c, false, false);

    // C/D layout: lane n = m (cols 0..3 useful); vgpr j -> row j + 8*kh.
    if (m < 4) {
#pragma unroll
        for (int j = 0; j < 8; ++j) {
            long r = r0 + 8 * kh + j;
            G0[r * 4 + m] = c[j] + bias;   // 4 lanes x contiguous floats -> 16B
        }
    }
}

// ---------------------------------------------------------------------------
// Activations via the CDNA5 TRANS-unit TANH op: one v_tanh_f32 per activation
// instead of v_exp_f32 + v_rcp_f32, halving the dependent-TRANS depth of the
// serial scan (the latency-bound kernel: 256 waves, 512 dependent steps).
// ---------------------------------------------------------------------------
__device__ __forceinline__ float tanh_hw(float x) {
#if __has_builtin(__builtin_amdgcn_tanhf)
    return __builtin_amdgcn_tanhf(x);
#else
    float r;
    asm("v_tanh_f32 %0, %1" : "=v"(r) : "v"(x));
    return r;
#endif
}
__device__ __forceinline__ float sigf(float x) {
    return fmaf(0.5f, tanh_hw(0.5f * x), 0.5f);   // sigmoid(x)=0.5*tanh(x/2)+0.5
}

// ---------------------------------------------------------------------------
// Kernel B: fused 5-layer scan. One thread per batch element; all (h,c) state
// and the tiny per-layer weights live in registers. Reads G0 in 128B
// per-thread chunks (8 steps) so every fetched line is fully consumed.
// ---------------------------------------------------------------------------
__global__ __launch_bounds__(128) void lstm_scan_fused(
    const float* __restrict__ G0,    // [B,T,4] precomputed layer-0 gates
    const float* __restrict__ Whh0,  // [4,1]
    const float* __restrict__ WihL,  // [L-1,4,1]
    const float* __restrict__ WhhL,  // [L-1,4,1]
    const float* __restrict__ bihL,  // [L-1,4]
    const float* __restrict__ bhhL,  // [L-1,4]
    const float* __restrict__ fcw, const float* __restrict__ fcb,
    const float* __restrict__ fc2w, const float* __restrict__ fc2b,
    float* __restrict__ out)         // [B]
{
    const int b = blockIdx.x * blockDim.x + threadIdx.x;
    if (b >= BSZ) return;

    float whh0[4], wih[NLAY - 1][4], whh[NLAY - 1][4], bl[NLAY - 1][4];
#pragma unroll
    for (int n = 0; n < 4; ++n) whh0[n] = Whh0[n];
#pragma unroll
    for (int l = 0; l < NLAY - 1; ++l)
#pragma unroll
        for (int n = 0; n < 4; ++n) {
            wih[l][n] = WihL[l * 4 + n];
            whh[l][n] = WhhL[l * 4 + n];
            bl[l][n]  = bihL[l * 4 + n] + bhhL[l * 4 + n];
        }

    float h[NLAY], cs[NLAY];
#pragma unroll
    for (int l = 0; l < NLAY; ++l) { h[l] = 0.f; cs[l] = 0.f; }

    const float4* gp = (const float4*)(G0 + (long)b * TLEN * 4);

    for (int t0 = 0; t0 < TLEN; t0 += 8) {
        if (t0 + 16 <= TLEN)
            __builtin_prefetch((const void*)(gp + t0 + 8), 0, 1); // global_prefetch

        float4 gb[8];
#pragma unroll
        for (int j = 0; j < 8; ++j) gb[j] = gp[t0 + j];  // 128B contiguous/lane

#pragma unroll
        for (int j = 0; j < 8; ++j) {
            // layer 0: x-projection + bias already in gb; add recurrent term
            float gi = gb[j].x + whh0[0] * h[0];
            float gf = gb[j].y + whh0[1] * h[0];
            float gg = gb[j].z + whh0[2] * h[0];
            float go = gb[j].w + whh0[3] * h[0];
            cs[0] = sigf(gf) * cs[0] + sigf(gi) * tanh_hw(gg);
            h[0]  = sigf(go) * tanh_hw(cs[0]);

#pragma unroll
            for (int l = 1; l < NLAY; ++l) {
                float xin = h[l - 1];
                float i2 = wih[l-1][0] * xin + whh[l-1][0] * h[l] + bl[l-1][0];
                float f2 = wih[l-1][1] * xin + whh[l-1][1] * h[l] + bl[l-1][1];
                float g2 = wih[l-1][2] * xin + whh[l-1][2] * h[l] + bl[l-1][2];
                float o2 = wih[l-1][3] * xin + whh[l-1][3] * h[l] + bl[l-1][3];
                cs[l] = sigf(f2) * cs[l] + sigf(i2) * tanh_hw(g2);
                h[l]  = sigf(o2) * tanh_hw(cs[l]);
            }
        }
    }

    // head: relu(fc) -> fc (again, per reference) -> fc2   (all 1x1 scalars)
    float fw = fcw[0], fb = fcb[0];
    float v = h[NLAY - 1];
    v = fmaxf(v * fw + fb, 0.f);
    v = v * fw + fb;
    v = v * fc2w[0] + fc2b[0];
    out[b] = v;
}

// ---------------------------------------------------------------------------
extern "C" void kernel_launch(void* const* d_in, const int* in_sizes, int n_in,
                              void* d_out, int out_size, void* d_ws, size_t ws_size,
                              hipStream_t stream) {
    const float* x     = (const float*)d_in[0];   // [B,T,F]
    const float* Wih0  = (const float*)d_in[1];   // [4,8]
    const float* Whh0  = (const float*)d_in[2];   // [4,1]
    const float* bih0  = (const float*)d_in[3];   // [4]
    const float* bhh0  = (const float*)d_in[4];   // [4]
    const float* WihL  = (const float*)d_in[5];   // [4,4,1]
    const float* WhhL  = (const float*)d_in[6];   // [4,4,1]
    const float* bihL  = (const float*)d_in[7];   // [4,4]
    const float* bhhL  = (const float*)d_in[8];   // [4,4]
    const float* fcw   = (const float*)d_in[9];
    const float* fcb   = (const float*)d_in[10];
    const float* fc2w  = (const float*)d_in[11];
    const float* fc2b  = (const float*)d_in[12];
    float* out = (float*)d_out;

    float* G0 = (float*)d_ws;  // [B,T,4] f32 = 64 MB

    // Kernel A: 262144 tiles of 16 rows, 8 waves (256 thr) per block
    const int tiles  = (BSZ * TLEN) / 16;
    const int blkA   = tiles / 8;            // 32768
    lstm0_gates_wmma<<<blkA, 256, 0, stream>>>(x, Wih0, bih0, bhh0, G0);

    // Kernel B: one thread per batch element
    lstm_scan_fused<<<BSZ / 128, 128, 0, stream>>>(
        G0, Whh0, WihL, WhhL, bihL, bhhL, fcw, fcb, fc2w, fc2b, out);
}